// CountSketch_85710367359545
// MI455X (gfx1250) — compile-verified
//
#include <hip/hip_runtime.h>
#include <stdint.h>

#define D_IN  16384
#define D_F   1024
#define BATCH 4096
#define ROWS_PER_BLOCK 8
#define THREADS 256

typedef float v4f __attribute__((ext_vector_type(4)));
typedef int   v4i __attribute__((ext_vector_type(4)));

// Prologue: fuse (bucket index | sign bit of s_hash) into one int per column.
__global__ __launch_bounds__(THREADS) void cs_encode(const float* __restrict__ s_hash,
                                                     const int* __restrict__ i_hash,
                                                     int* __restrict__ codes) {
    int j = blockIdx.x * THREADS + threadIdx.x;
    if (j < D_IN) {
        uint32_t sign = __float_as_uint(s_hash[j]) & 0x80000000u;
        codes[j] = (int)(((uint32_t)i_hash[j] & (uint32_t)(D_F - 1)) | sign);
    }
}

__global__ __launch_bounds__(THREADS) void cs_scatter(const float* __restrict__ x,
                                                      const int* __restrict__ codes,
                                                      float* __restrict__ out) {
    __shared__ int   lds_codes[D_IN];               // 64 KB
    __shared__ float accum[ROWS_PER_BLOCK][D_F];    // 32 KB  (96 KB total; 2 blocks/WGP)

    const int tid  = threadIdx.x;
    const int wave = tid >> 5;     // wave32: 8 waves per block
    const int lane = tid & 31;
    const int row0 = blockIdx.x * ROWS_PER_BLOCK;

    // (1) Stage the 64 KB code table global->LDS with gfx1250 async copies
    //     (no VGPR staging, tracked by ASYNCcnt).
    {
        const uint32_t lds_base = (uint32_t)(uintptr_t)&lds_codes[0];
        const uint64_t gbase    = (uint64_t)(uintptr_t)codes;
        #pragma unroll
        for (int i = 0; i < D_IN / (THREADS * 4); ++i) {   // 16 x b128 per thread
            const uint32_t byte_off = (uint32_t)(i * THREADS + tid) * 16u;
            uint32_t laddr = lds_base + byte_off;
            uint64_t gaddr = gbase + byte_off;
            asm volatile("global_load_async_to_lds_b128 %0, %1, off"
                         :: "v"(laddr), "v"(gaddr) : "memory");
        }
    }

    // (2) Zero bucket accumulators while the async copies are in flight.
    #pragma unroll
    for (int i = tid; i < ROWS_PER_BLOCK * D_F; i += THREADS)
        (&accum[0][0])[i] = 0.0f;

    asm volatile("s_wait_asynccnt 0x0" ::: "memory");
    __syncthreads();

    // (3) One wave per batch row: stream x non-temporally, scatter via ds_add_f32.
    const float* xrow = x + (size_t)(row0 + wave) * D_IN;
    float*       arow = &accum[wave][0];
    #pragma unroll 4
    for (int it = 0; it < D_IN / 128; ++it) {
        const int j = it * 128 + lane * 4;                      // coalesced b128
        v4f v = __builtin_nontemporal_load((const v4f*)(xrow + j));
        v4i c = *(const v4i*)(&lds_codes[j]);                   // ds_load_b128
        atomicAdd(&arow[c.x & (D_F - 1)],
                  __uint_as_float(__float_as_uint(v.x) ^ ((uint32_t)c.x & 0x80000000u)));
        atomicAdd(&arow[c.y & (D_F - 1)],
                  __uint_as_float(__float_as_uint(v.y) ^ ((uint32_t)c.y & 0x80000000u)));
        atomicAdd(&arow[c.z & (D_F - 1)],
                  __uint_as_float(__float_as_uint(v.z) ^ ((uint32_t)c.z & 0x80000000u)));
        atomicAdd(&arow[c.w & (D_F - 1)],
                  __uint_as_float(__float_as_uint(v.w) ^ ((uint32_t)c.w & 0x80000000u)));
    }

    __syncthreads();

    // (4) Flush 8 rows x 1024 floats with coalesced 128-bit NT stores.
    #pragma unroll
    for (int i = tid; i < ROWS_PER_BLOCK * (D_F / 4); i += THREADS) {
        const int r  = i >> 8;        // / 256 float4s per row
        const int c4 = i & 255;
        v4f v = *(const v4f*)&accum[r][c4 * 4];
        __builtin_nontemporal_store(v, (v4f*)(out + (size_t)(row0 + r) * D_F + c4 * 4));
    }
}

extern "C" void kernel_launch(void* const* d_in, const int* in_sizes, int n_in,
                              void* d_out, int out_size, void* d_ws, size_t ws_size,
                              hipStream_t stream) {
    (void)in_sizes; (void)n_in; (void)out_size; (void)ws_size;
    const float* x      = (const float*)d_in[0];
    const float* s_hash = (const float*)d_in[1];
    const int*   i_hash = (const int*)d_in[2];
    float* out   = (float*)d_out;
    int*   codes = (int*)d_ws;      // 64 KB scratch for fused codes

    cs_encode<<<D_IN / THREADS, THREADS, 0, stream>>>(s_hash, i_hash, codes);
    cs_scatter<<<BATCH / ROWS_PER_BLOCK, THREADS, 0, stream>>>(x, codes, out);
}